// HGTLayer_68204080660553
// MI455X (gfx1250) — compile-verified
//
#include <hip/hip_runtime.h>
#include <hip/hip_bf16.h>

#define Dn 128
#define Hh 4

typedef __attribute__((ext_vector_type(2))) float v2f;
typedef __attribute__((ext_vector_type(8))) float v8f;

// ---- order-preserving float<->uint encoding for atomic segment-max ----
__device__ __forceinline__ unsigned encf(float f) {
    unsigned u = __float_as_uint(f);
    return (u & 0x80000000u) ? ~u : (u | 0x80000000u);
}
__device__ __forceinline__ float decf(unsigned u) {
    return (u & 0x80000000u) ? __uint_as_float(u & 0x7FFFFFFFu)
                             : __uint_as_float(~u);
}

// =====================================================================
// C[M x 128] = A[M x 128] @ W[128 x 128] (+ bias) (+ addend)   via WMMA
// 256 threads = 8 waves; each wave computes a 16-row tile.
// Weights staged in LDS (64KB) once per block, reused by all waves.
// =====================================================================
__global__ __launch_bounds__(256) void gemm128_wmma(
    const float* __restrict__ A, const float* __restrict__ W,
    const float* __restrict__ bias, const float* __restrict__ addend,
    float* __restrict__ Out, int M)
{
    __shared__ float sW[Dn * Dn];  // 64 KB

    // cooperative stage of W into LDS: 4096 float4 / 256 threads = 16 each
    {
        const float4* W4 = (const float4*)W;
        float4* sW4 = (float4*)sW;
        int t = threadIdx.x;
#pragma unroll
        for (int i = 0; i < 16; ++i) sW4[t + i * 256] = W4[t + i * 256];
    }
    __syncthreads();

    const int wave  = threadIdx.x >> 5;
    const int lane  = threadIdx.x & 31;
    const int row16 = (blockIdx.x * 8 + wave) * 16;
    const int r     = lane & 15;
    const int khalf = lane >> 4;        // 0 or 1

    // A fragments for the whole K=128: 32 steps of K=4, v2f per lane
    int arow = row16 + r;
    if (arow >= M) arow = M - 1;        // clamp (EXEC stays all-1 for WMMA)
    const float* Arow = A + (size_t)arow * Dn;
    v2f afrag[32];
#pragma unroll
    for (int k = 0; k < 32; ++k)
        afrag[k] = *(const v2f*)(Arow + k * 4 + khalf * 2);

    const int col = lane & 15;
#pragma unroll 1
    for (int n = 0; n < 8; ++n) {
        const int c = n * 16 + col;
        v8f acc = {};
#pragma unroll
        for (int k = 0; k < 32; ++k) {
            const int kk = k * 4 + khalf * 2;
            v2f b;
            b.x = sW[kk * Dn + c];
            b.y = sW[(kk + 1) * Dn + c];
            acc = __builtin_amdgcn_wmma_f32_16x16x4_f32(
                false, afrag[k], false, b, (short)0, acc, false, false);
        }
        const float bv = bias ? bias[c] : 0.0f;
#pragma unroll
        for (int v = 0; v < 8; ++v) {
            const int rr = row16 + v + khalf * 8;
            if (rr < M) {
                float val = acc[v] + bv;
                if (addend) val += addend[(size_t)rr * Dn + c];
                Out[(size_t)rr * Dn + c] = val;
            }
        }
    }
}

// =====================================================================
// Pass 1: per-edge attention scores + atomic segment-max (wave per edge)
// =====================================================================
__global__ __launch_bounds__(256) void edge_scores(
    const float* __restrict__ K, const float* __restrict__ Q,
    const int* __restrict__ src, const int* __restrict__ dst,
    float* __restrict__ scores, unsigned* __restrict__ m_enc, int E)
{
    const int e    = blockIdx.x * 8 + (threadIdx.x >> 5);
    const int lane = threadIdx.x & 31;
    if (e >= E) return;
    const int s = src[e], d = dst[e];
    const float4 q = *(const float4*)(Q + (size_t)d * Dn + lane * 4);
    const float4 k = *(const float4*)(K + (size_t)s * Dn + lane * 4);
    float p = q.x * k.x + q.y * k.y + q.z * k.z + q.w * k.w;
    // reduce the 8 lanes of each head (head = lane>>3)
    p += __shfl_xor(p, 1, 32);
    p += __shfl_xor(p, 2, 32);
    p += __shfl_xor(p, 4, 32);
    const float sc = p * 0.17677669529663687f;  // 1/sqrt(32)
    const int head = lane >> 3;
    if ((lane & 7) == 0) {
        scores[(size_t)e * Hh + head] = sc;
        atomicMax(&m_enc[(size_t)d * Hh + head], encf(sc));
    }
}

// =====================================================================
// Pass 2: ex = exp(score - m[dst]); segment-sum via fp32 hw atomics
// thread per (edge, head); scores buffer updated in place to hold ex
// =====================================================================
__global__ __launch_bounds__(256) void edge_exp(
    float* __restrict__ ex, const unsigned* __restrict__ m_enc,
    float* __restrict__ ssum, const int* __restrict__ dst, int EH)
{
    const int i = blockIdx.x * 256 + threadIdx.x;
    if (i >= EH) return;
    const int e = i >> 2, h = i & 3;
    const int d = dst[e];
    const float mv = decf(m_enc[(size_t)d * Hh + h]);
    const float v  = __expf(ex[i] - mv);
    ex[i] = v;
    unsafeAtomicAdd(&ssum[(size_t)d * Hh + h], v);
}

// =====================================================================
// Pass 3: scatter-add attn * V into agg[dst]   (wave per edge)
// =====================================================================
__global__ __launch_bounds__(256) void edge_agg(
    const float* __restrict__ V, const float* __restrict__ ex,
    const float* __restrict__ ssum, const int* __restrict__ src,
    const int* __restrict__ dst, float* __restrict__ agg, int E)
{
    const int e    = blockIdx.x * 8 + (threadIdx.x >> 5);
    const int lane = threadIdx.x & 31;
    if (e >= E) return;
    const int s = src[e], d = dst[e];
    const int head = lane >> 3;
    const float a = ex[(size_t)e * Hh + head] / ssum[(size_t)d * Hh + head];
    const float4 v = *(const float4*)(V + (size_t)s * Dn + lane * 4);
    float* p = agg + (size_t)d * Dn + lane * 4;
    unsafeAtomicAdd(p + 0, v.x * a);
    unsafeAtomicAdd(p + 1, v.y * a);
    unsafeAtomicAdd(p + 2, v.z * a);
    unsafeAtomicAdd(p + 3, v.w * a);
}

// =====================================================================
// LayerNorm + affine + ReLU, wave per row
// =====================================================================
__global__ __launch_bounds__(256) void ln_relu(
    const float* __restrict__ y, const float* __restrict__ g,
    const float* __restrict__ be, float* __restrict__ out, int Nr)
{
    const int row  = blockIdx.x * 8 + (threadIdx.x >> 5);
    const int lane = threadIdx.x & 31;
    if (row >= Nr) return;
    const float4 x = *(const float4*)(y + (size_t)row * Dn + lane * 4);
    float s1 = x.x + x.y + x.z + x.w;
    float s2 = x.x * x.x + x.y * x.y + x.z * x.z + x.w * x.w;
#pragma unroll
    for (int m = 1; m < 32; m <<= 1) {
        s1 += __shfl_xor(s1, m, 32);
        s2 += __shfl_xor(s2, m, 32);
    }
    const float mu  = s1 * (1.0f / 128.0f);
    const float var = s2 * (1.0f / 128.0f) - mu * mu;
    const float rs  = rsqrtf(var + 1e-5f);
    const float4 gg = *(const float4*)(g + lane * 4);
    const float4 bb = *(const float4*)(be + lane * 4);
    float4 o;
    o.x = fmaxf((x.x - mu) * rs * gg.x + bb.x, 0.0f);
    o.y = fmaxf((x.y - mu) * rs * gg.y + bb.y, 0.0f);
    o.z = fmaxf((x.z - mu) * rs * gg.z + bb.z, 0.0f);
    o.w = fmaxf((x.w - mu) * rs * gg.w + bb.w, 0.0f);
    *(float4*)(out + (size_t)row * Dn + lane * 4) = o;
}

extern "C" void kernel_launch(void* const* d_in, const int* in_sizes, int n_in,
                              void* d_out, int out_size, void* d_ws, size_t ws_size,
                              hipStream_t stream) {
    (void)in_sizes; (void)n_in; (void)out_size; (void)ws_size;
    const int N = 100000, E = 400000;
    const size_t ND = (size_t)N * Dn;
    const size_t NH = (size_t)N * Hh;
    const size_t EH = (size_t)E * Hh;

    const float* x_drug  = (const float*)d_in[0];
    const float* x_dis   = (const float*)d_in[1];
    const float* Wp_drug = (const float*)d_in[2];
    const float* bp_drug = (const float*)d_in[3];
    const float* Wp_dis  = (const float*)d_in[4];
    const float* bp_dis  = (const float*)d_in[5];
    const float* Wk_r0   = (const float*)d_in[6];
    const float* Wv_r0   = (const float*)d_in[7];
    const float* Wk_r1   = (const float*)d_in[8];
    const float* Wv_r1   = (const float*)d_in[9];
    const float* Wu_drug = (const float*)d_in[10];
    const float* bu_drug = (const float*)d_in[11];
    const float* Wu_dis  = (const float*)d_in[12];
    const float* bu_dis  = (const float*)d_in[13];
    const float* g_drug  = (const float*)d_in[14];
    const float* be_drug = (const float*)d_in[15];
    const float* g_dis   = (const float*)d_in[16];
    const float* be_dis  = (const float*)d_in[17];
    const int*   ei0     = (const int*)d_in[18];   // [2,E]: src drug, dst dis
    const int*   ei1     = (const int*)d_in[19];   // [2,E]: src dis,  dst drug
    float* out = (float*)d_out;

    // ---- workspace carve-up ----
    float* w = (float*)d_ws;
    float* h_drug   = w; w += ND;
    float* h_dis    = w; w += ND;
    float* K0       = w; w += ND;   // later reused as y_drug (pre-LN)
    float* V0       = w; w += ND;
    float* K1       = w; w += ND;   // later reused as y_dis
    float* V1       = w; w += ND;
    float* agg_dis  = w; w += ND;   // --- zeroed region starts here ---
    float* agg_drug = w; w += ND;
    unsigned* m0    = (unsigned*)w; w += NH;
    float* ss0      = w; w += NH;
    unsigned* m1    = (unsigned*)w; w += NH;
    float* ss1      = w; w += NH;   // --- zeroed region ends here ---
    float* ex0      = w; w += EH;
    float* ex1      = w; w += EH;

    hipMemsetAsync(agg_dis, 0, (2 * ND + 4 * NH) * sizeof(float), stream);

    const dim3 blk(256);
    const int gbG = (N + 127) / 128;        // gemm: 128 rows/block
    const int gbE = (E + 7) / 8;            // wave-per-edge
    const int gbH = ((int)(E * Hh) + 255) / 256;
    const int gbR = (N + 7) / 8;            // wave-per-row

    // node-level projections
    gemm128_wmma<<<gbG, blk, 0, stream>>>(x_drug, Wp_drug, bp_drug, nullptr, h_drug, N);
    gemm128_wmma<<<gbG, blk, 0, stream>>>(x_dis,  Wp_dis,  bp_dis,  nullptr, h_dis,  N);

    // node-level K/V (algebraically equal to per-edge K/V, 4x fewer FLOPs)
    gemm128_wmma<<<gbG, blk, 0, stream>>>(h_drug, Wk_r0, nullptr, nullptr, K0, N);
    gemm128_wmma<<<gbG, blk, 0, stream>>>(h_drug, Wv_r0, nullptr, nullptr, V0, N);
    gemm128_wmma<<<gbG, blk, 0, stream>>>(h_dis,  Wk_r1, nullptr, nullptr, K1, N);
    gemm128_wmma<<<gbG, blk, 0, stream>>>(h_dis,  Wv_r1, nullptr, nullptr, V1, N);

    // relation 0: drug -> dis (Q from h_dis)
    edge_scores<<<gbE, blk, 0, stream>>>(K0, h_dis, ei0, ei0 + E, ex0, m0, E);
    edge_exp   <<<gbH, blk, 0, stream>>>(ex0, m0, ss0, ei0 + E, (int)(E * Hh));
    edge_agg   <<<gbE, blk, 0, stream>>>(V0, ex0, ss0, ei0, ei0 + E, agg_dis, E);

    // relation 1: dis -> drug (Q from h_drug)
    edge_scores<<<gbE, blk, 0, stream>>>(K1, h_drug, ei1, ei1 + E, ex1, m1, E);
    edge_exp   <<<gbH, blk, 0, stream>>>(ex1, m1, ss1, ei1 + E, (int)(E * Hh));
    edge_agg   <<<gbE, blk, 0, stream>>>(V1, ex1, ss1, ei1, ei1 + E, agg_drug, E);

    // update: y = h + agg @ Wu + bu   (K0/K1 reused as y buffers)
    gemm128_wmma<<<gbG, blk, 0, stream>>>(agg_drug, Wu_drug, bu_drug, h_drug, K0, N);
    gemm128_wmma<<<gbG, blk, 0, stream>>>(agg_dis,  Wu_dis,  bu_dis,  h_dis,  K1, N);

    // layernorm + relu into stacked output [2, N, D]
    ln_relu<<<gbR, blk, 0, stream>>>(K0, g_drug, be_drug, out,      N);
    ln_relu<<<gbR, blk, 0, stream>>>(K1, g_dis,  be_dis,  out + ND, N);
}